// ValueNetwork_82463372083417
// MI455X (gfx1250) — compile-verified
//
#include <hip/hip_runtime.h>

typedef float v2f __attribute__((ext_vector_type(2)));
typedef float v8f __attribute__((ext_vector_type(8)));

#define IN_CH 128
#define HID   128

// ---------------- degree / normalization ----------------

__global__ __launch_bounds__(256) void k_init_deg(float* __restrict__ deg, int n) {
    int i = blockIdx.x * 256 + threadIdx.x;
    if (i < n) deg[i] = 1.0f;                      // self-loop weight 1
}

__global__ __launch_bounds__(256) void k_deg_atomic(const long long* __restrict__ dst,
                                                    const float* __restrict__ ew,
                                                    float* __restrict__ deg, int E) {
    int e = blockIdx.x * 256 + threadIdx.x;
    if (e < E) {
        long long d = __builtin_nontemporal_load(dst + e);   // streaming: keep L2 for features
        float     w = __builtin_nontemporal_load(ew + e);
        atomicAdd(deg + (int)d, w);
    }
}

__global__ __launch_bounds__(256) void k_rsqrt(float* __restrict__ deg, int n) {
    int i = blockIdx.x * 256 + threadIdx.x;
    if (i < n) {
        float d = deg[i];
        deg[i] = (d > 0.0f) ? rsqrtf(d) : 0.0f;    // deg -> dinv in place
    }
}

// ---- one-shot weight swizzle: Ws[(k>>1)*256 + col*2 + (k&1)] = W[k*128 + col] ----
// Puts the (k, k+1) pair a lane needs for a WMMA B fragment at adjacent addresses,
// so the GEMM's LDS reads become single aligned ds_load_b64s.
__global__ __launch_bounds__(256) void k_swizzle_w(const float* __restrict__ W,
                                                   float* __restrict__ Ws) {
    int t = blockIdx.x * 256 + threadIdx.x;        // 0 .. 16383
    int k = t >> 7, col = t & 127;
    Ws[((k >> 1) * HID + col) * 2 + (k & 1)] = W[t];
}

// ---- fused dense transform:  H = act(A) @ W ;  Craw = H ; Cself = dinv^2 * H ----
// act(A) = relu(A + bias) when RELU, else identity (applies PREVIOUS layer's bias+relu).
// Swizzled W is staged block-wide into LDS with gfx1250 async-to-LDS DMA; B fragments
// then come from single ds_load_b64s. One wave per 16-row slab, 8 v8f accumulators.
// V_WMMA_F32_16X16X4_F32 layouts (ISA 7.12.2):
//   A 16x4:  lanes 0-15: v0=K0,v1=K1 ; lanes 16-31: v0=K2,v1=K3   (M = lane&15)
//   B 4x16:  v0: K0 (lanes 0-15) / K2 (lanes 16-31) ; v1: K1 / K3 (N = lane&15)
//   C 16x16: VGPR v: lanes 0-15 -> M=v, lanes 16-31 -> M=v+8, N = lane&15
template <bool RELU>
__global__ __launch_bounds__(128) void k_gemm_fused(const float* A,
                                                    const float* __restrict__ Wswz,
                                                    const float* __restrict__ bias,
                                                    const float* __restrict__ dinv,
                                                    float* __restrict__ Craw,
                                                    float* Cself, int M) {
    __shared__ float Wlds[HID * HID];   // 64 KB, swizzled layout
    __shared__ float blds[HID];

    const int t = threadIdx.x;

    // ---- async DMA stage of swizzled W into LDS: 128 threads x 32 x b128 = 64 KB ----
    {
        unsigned           lds = (unsigned)(size_t)(&Wlds[0]) + t * 16u;
        unsigned long long ga  = (unsigned long long)(size_t)Wswz + t * 16ull;
#pragma unroll
        for (int i = 0; i < 32; ++i) {
            asm volatile("global_load_async_to_lds_b128 %0, %1, off offset:%2"
                         :: "v"(lds), "v"(ga), "i"(i * 2048) : "memory");
        }
    }
    if (RELU && t < HID / 4) {
        ((float4*)blds)[t] = ((const float4*)bias)[t];
    }
    asm volatile("s_wait_asynccnt 0x0" ::: "memory");
    __syncthreads();

    const int wave  = t >> 5;
    const int lane  = t & 31;
    const int tile  = blockIdx.x * 4 + wave;
    if (tile * 16 >= M) return;
    const int half  = lane >> 4;            // 0 or 1
    const int l15   = lane & 15;
    const int kBase = half << 1;            // 0 or 2

    const float* Arow = A + (size_t)(tile * 16 + l15) * IN_CH;
    v8f acc[8] = {};

    for (int k0 = 0; k0 < IN_CH; k0 += 4) {
        v2f a = *(const v2f*)(Arow + k0 + kBase);      // K = kBase, kBase+1
        if (RELU) {
            a.x = fmaxf(a.x + blds[k0 + kBase + 0], 0.0f);
            a.y = fmaxf(a.y + blds[k0 + kBase + 1], 0.0f);
        }
        // swizzled: pair (k0+kBase, k0+kBase+1) at [(k0/2 + half)*256 + col*2]
        const float* Wp = &Wlds[(k0 / 2 + half) * (2 * HID) + l15 * 2];
#pragma unroll
        for (int n = 0; n < 8; ++n) {
            v2f b = *(const v2f*)(Wp + n * 32);        // single ds_load_b64
            acc[n] = __builtin_amdgcn_wmma_f32_16x16x4_f32(
                false, a, false, b, (short)0, acc[n], false, false);
        }
    }

    // epilogue: write raw H and self-loop term dinv^2 * H
    float dv2[8];
#pragma unroll
    for (int v = 0; v < 8; ++v) {
        float s = dinv[tile * 16 + v + half * 8];
        dv2[v] = s * s;
    }
#pragma unroll
    for (int n = 0; n < 8; ++n) {
#pragma unroll
        for (int v = 0; v < 8; ++v) {
            size_t idx = (size_t)(tile * 16 + v + half * 8) * HID + n * 16 + l15;
            float h = acc[n][v];
            Craw[idx]  = h;
            Cself[idx] = dv2[v] * h;
        }
    }
}

// ---------------- edge scatter: out[dst] += dinv[s]*w*dinv[d] * h[src] ----------------
// one wave per edge: 32 lanes x float4 = 128 channels, coalesced 512B gather
__global__ __launch_bounds__(256) void k_edge_agg(const long long* __restrict__ src,
                                                  const long long* __restrict__ dst,
                                                  const float* __restrict__ ew,
                                                  const float* __restrict__ dinv,
                                                  const float* __restrict__ h,
                                                  float* __restrict__ out, int E) {
    long long t = (long long)blockIdx.x * 256 + threadIdx.x;
    long long e = t >> 5;
    if (e >= E) return;
    int lane = (int)(t & 31);
    int s = (int)__builtin_nontemporal_load(src + e);   // streaming edge lists: NT
    int d = (int)__builtin_nontemporal_load(dst + e);
    float w = __builtin_nontemporal_load(ew + e);
    float norm = dinv[s] * w * dinv[d];
    float4 v = ((const float4*)(h + (size_t)s * HID))[lane];   // L2-resident gather
    float* op = out + (size_t)d * HID + lane * 4;
    atomicAdd(op + 0, norm * v.x);
    atomicAdd(op + 1, norm * v.y);
    atomicAdd(op + 2, norm * v.z);
    atomicAdd(op + 3, norm * v.w);
}

// ------- head: value[i] = relu(agg2[i,:]+b2)·Wfc[0:128] + action[i]*Wfc[128] + bfc -------
__global__ __launch_bounds__(256) void k_final(const float* __restrict__ h,
                                               const float* __restrict__ b2,
                                               const float* __restrict__ action,
                                               const float* __restrict__ Wfc,
                                               const float* __restrict__ bfc,
                                               float* __restrict__ out, int N) {
    int t = blockIdx.x * 256 + threadIdx.x;
    int i = t >> 5;
    int lane = t & 31;
    if (i >= N) return;
    float4 hv = ((const float4*)(h + (size_t)i * HID))[lane];
    float4 bb = ((const float4*)b2)[lane];
    float4 wv = ((const float4*)Wfc)[lane];
    float s = fmaxf(hv.x + bb.x, 0.0f) * wv.x
            + fmaxf(hv.y + bb.y, 0.0f) * wv.y
            + fmaxf(hv.z + bb.z, 0.0f) * wv.z
            + fmaxf(hv.w + bb.w, 0.0f) * wv.w;
#pragma unroll
    for (int off = 16; off > 0; off >>= 1)
        s += __shfl_down(s, off, 32);
    if (lane == 0)
        out[i] = s + action[i] * Wfc[HID] + bfc[0];
}

// ---------------- driver ----------------

extern "C" void kernel_launch(void* const* d_in, const int* in_sizes, int n_in,
                              void* d_out, int out_size, void* d_ws, size_t ws_size,
                              hipStream_t stream) {
    const float*     x      = (const float*)d_in[0];
    const long long* ei     = (const long long*)d_in[1];   // int64 [2, E]
    const float*     ew     = (const float*)d_in[2];
    const float*     action = (const float*)d_in[3];
    const float*     W1     = (const float*)d_in[4];
    const float*     b1     = (const float*)d_in[5];
    const float*     W2     = (const float*)d_in[6];
    const float*     b2     = (const float*)d_in[7];
    const float*     Wfc    = (const float*)d_in[8];
    const float*     bfc    = (const float*)d_in[9];
    float*           out    = (float*)d_out;

    const int N = in_sizes[0] / IN_CH;      // 100000
    const int E = in_sizes[1] / 2;          // 3200000
    const long long* src = ei;
    const long long* dst = ei + E;

    // workspace layout: [dinv (400KB) | W1s | W2s | bufA (51.2MB) | bufB (51.2MB)]
    char*  ws   = (char*)d_ws;
    float* dinv = (float*)ws;
    float* W1s  = (float*)(ws + (size_t)409600);
    float* W2s  = (float*)(ws + (size_t)409600 + 65536);
    float* bufA = (float*)(ws + (size_t)(1 << 20));
    float* bufB = (float*)(ws + (size_t)(1 << 20) + (size_t)52428800);

    const int nb  = (N + 255) / 256;
    const int eb  = (E + 255) / 256;
    const int evb = (int)(((long long)E * 32 + 255) / 256);
    const int gb  = ((N + 15) / 16 + 3) / 4;
    const int fb  = (N * 32 + 255) / 256;

    // weight swizzles (64 KB each, one-shot)
    k_swizzle_w      <<<64, 256, 0, stream>>>(W1, W1s);
    k_swizzle_w      <<<64, 256, 0, stream>>>(W2, W2s);

    // normalization coefficients: deg -> dinv (in place)
    k_init_deg       <<<nb,  256, 0, stream>>>(dinv, N);
    k_deg_atomic     <<<eb,  256, 0, stream>>>(dst, ew, dinv, E);
    k_rsqrt          <<<nb,  256, 0, stream>>>(dinv, N);

    // layer 1: bufA = x@W1 (raw), bufB = dinv^2 * (x@W1); then edge scatter into bufB
    k_gemm_fused<false><<<gb, 128, 0, stream>>>(x, W1s, nullptr, dinv, bufA, bufB, N);
    k_edge_agg       <<<evb, 256, 0, stream>>>(src, dst, ew, dinv, bufA, bufB, E);

    // layer 2: reads bufB through relu(.+b1); bufA = h2 raw, bufB = dinv^2 * h2
    // (Cself rows are wave-exclusive with the rows read from A: safe in-place)
    k_gemm_fused<true> <<<gb, 128, 0, stream>>>(bufB, W2s, b1, dinv, bufA, bufB, N);
    k_edge_agg       <<<evb, 256, 0, stream>>>(src, dst, ew, dinv, bufA, bufB, E);

    // head: fused relu(.+b2) dot Wfc + action*Wfc[128] + bfc
    k_final          <<<fb,  256, 0, stream>>>(bufB, b2, action, Wfc, bfc, out, N);
}